// GCN_45672682226339
// MI455X (gfx1250) — compile-verified
//
#include <hip/hip_runtime.h>

typedef float v2f __attribute__((ext_vector_type(2)));
typedef float v8f __attribute__((ext_vector_type(8)));

#define F_IN 128
#define HID  128
#define HID2 64
#define OUTC 32

static inline int ceil_div(int a, int b) { return (a + b - 1) / b; }

// ---------------- degree / normalization ----------------

__global__ void deg_init_kernel(float* __restrict__ deg, int N) {
  int i = blockIdx.x * blockDim.x + threadIdx.x;
  if (i < N) deg[i] = 1.0f;  // self-loop contributes 1
}

__global__ void deg_accum_kernel(const int* __restrict__ dst, float* __restrict__ deg, int E) {
  int e = blockIdx.x * blockDim.x + threadIdx.x;
  if (e < E) atomicAdd(&deg[dst[e]], 1.0f);
}

__global__ void deg_rsqrt_kernel(float* __restrict__ deg, int N) {
  int i = blockIdx.x * blockDim.x + threadIdx.x;
  if (i < N) deg[i] = rsqrtf(deg[i]);  // deg >= 1 guaranteed (self-loops)
}

// ---------------- dense transform: C[MxNcol] = A[MxK] @ B[KxNcol] ----------------
// f32 WMMA 16x16x4, one wave per 16x16 output tile, 4 waves per block.
// A-matrix layout (32-bit 16x4): lanes 0-15 hold M=lane with K=0 (v0), K=1 (v1);
// lanes 16-31 hold M=lane-16 with K=2 (v0), K=3 (v1). B symmetric over N.
// C/D: VGPR r -> M=r (lanes 0-15), M=r+8 (lanes 16-31); N = lane&15.
__global__ __launch_bounds__(128) void gemm_wmma_kernel(
    const float* __restrict__ A, const float* __restrict__ B, float* __restrict__ C,
    int M, int K, int Ncol) {
  int lane = threadIdx.x & 31;
  int wave = threadIdx.x >> 5;
  int mtile = blockIdx.x * 4 + wave;
  int row0 = mtile << 4;
  if (row0 >= M) return;             // wave-uniform exit: EXEC stays all-1s for WMMA
  int col0 = blockIdx.y << 4;
  int half = lane >> 4;              // 0: K=0,1   1: K=2,3
  int l16  = lane & 15;

  v8f acc = {0.f, 0.f, 0.f, 0.f, 0.f, 0.f, 0.f, 0.f};
  const float* arow = A + (size_t)(row0 + l16) * K;
  const float* bcol = B + col0 + l16;

  for (int k = 0; k < K; k += 4) {
    int ka = k + half * 2;
    v2f a, b;
    a.x = arow[ka + 0];
    a.y = arow[ka + 1];
    b.x = bcol[(size_t)(ka + 0) * Ncol];
    b.y = bcol[(size_t)(ka + 1) * Ncol];
    acc = __builtin_amdgcn_wmma_f32_16x16x4_f32(
        /*neg_a=*/false, a, /*neg_b=*/false, b,
        /*c_mod=*/(short)0, acc, /*reuse_a=*/false, /*reuse_b=*/false);
  }

  float* crow = C + (size_t)(row0 + half * 8) * Ncol + col0 + l16;
#pragma unroll
  for (int r = 0; r < 8; ++r)
    crow[(size_t)r * Ncol] = acc[r];
}

// ---------------- message passing ----------------

// acc[i][f] = XW[i][f] * dis[i]^2   (self-loop term; also serves as accumulator init)
__global__ void selfloop_init_kernel(const float* __restrict__ XW, const float* __restrict__ dis,
                                     float* __restrict__ acc, int N, int F) {
  int i = blockIdx.x * blockDim.x + threadIdx.x;
  if (i >= N * F) return;
  float d = dis[i / F];
  acc[i] = XW[i] * (d * d);
}

// one wave per edge; lane handles a float4 chunk of the feature vector
__global__ void scatter_kernel(const float* __restrict__ XW, const int* __restrict__ src,
                               const int* __restrict__ dst, const float* __restrict__ dis,
                               float* __restrict__ acc, int E, int F) {
  int gid = blockIdx.x * blockDim.x + threadIdx.x;
  int lane = gid & 31;
  int e = gid >> 5;
  if (e >= E) return;
  int s = src[e], d = dst[e];
  float nrm = dis[s] * dis[d];
  int nchunks = F >> 2;             // F = 128/64/32 -> 32/16/8 chunks
  if (lane < nchunks) {
    const float4 v = *((const float4*)(XW + (size_t)s * F) + lane);
    float* ap = acc + (size_t)d * F + lane * 4;
    atomicAdd(ap + 0, v.x * nrm);
    atomicAdd(ap + 1, v.y * nrm);
    atomicAdd(ap + 2, v.z * nrm);
    atomicAdd(ap + 3, v.w * nrm);
  }
}

// h = relu(h + bias)   (in place; F is a power of two)
__global__ void bias_relu_kernel(float* __restrict__ h, const float* __restrict__ b,
                                 int N, int F) {
  int i = blockIdx.x * blockDim.x + threadIdx.x;
  if (i >= N * F) return;
  float v = h[i] + b[i & (F - 1)];
  h[i] = v > 0.f ? v : 0.f;
}

// ---------------- pooling ----------------

__global__ void pool_count_kernel(const int* __restrict__ batch, float* __restrict__ cnt, int N) {
  int i = blockIdx.x * blockDim.x + threadIdx.x;
  if (i < N) atomicAdd(&cnt[batch[i]], 1.0f);
}

// one wave per node, 32 lanes == OUTC features
__global__ void pool_sum_kernel(const float* __restrict__ h, const int* __restrict__ batch,
                                float* __restrict__ out, int N) {
  int gid = blockIdx.x * blockDim.x + threadIdx.x;
  int lane = gid & 31;
  int node = gid >> 5;
  if (node >= N) return;
  int g = batch[node];
  atomicAdd(&out[(size_t)g * OUTC + lane], h[(size_t)node * OUTC + lane]);
}

__global__ void pool_div_kernel(float* __restrict__ out, const float* __restrict__ cnt, int G) {
  int i = blockIdx.x * blockDim.x + threadIdx.x;
  if (i >= G * OUTC) return;
  out[i] /= fmaxf(cnt[i >> 5], 1.0f);   // OUTC == 32
}

// ---------------- driver ----------------

extern "C" void kernel_launch(void* const* d_in, const int* in_sizes, int n_in,
                              void* d_out, int out_size, void* d_ws, size_t ws_size,
                              hipStream_t stream) {
  const float* x     = (const float*)d_in[0];
  const int*   ei    = (const int*)d_in[1];
  const int*   batch = (const int*)d_in[2];
  const float* W1    = (const float*)d_in[3];
  const float* b1    = (const float*)d_in[4];
  const float* W2    = (const float*)d_in[5];
  const float* b2    = (const float*)d_in[6];
  const float* W3    = (const float*)d_in[7];
  const float* b3    = (const float*)d_in[8];

  const int N = in_sizes[0] / F_IN;
  const int E = in_sizes[1] / 2;
  const int G = out_size / OUTC;
  const int* src = ei;
  const int* dst = ei + E;
  float* out = (float*)d_out;

  // workspace layout (floats): dis[N] | cnt[G] | bufA[N*HID] | bufB[N*HID]
  float* dis = (float*)d_ws;
  float* cnt = dis + N;
  size_t off = (size_t)N + (size_t)G;
  off = (off + 63) & ~(size_t)63;
  float* bufA = (float*)d_ws + off;             // GEMM output (messages)
  float* bufB = bufA + (size_t)N * HID;         // accumulator / hidden state

  const int TB = 256;

  hipMemsetAsync(cnt, 0, (size_t)G * sizeof(float), stream);
  hipMemsetAsync(out, 0, (size_t)out_size * sizeof(float), stream);

  // normalization coefficients
  deg_init_kernel<<<ceil_div(N, TB), TB, 0, stream>>>(dis, N);
  deg_accum_kernel<<<ceil_div(E, TB), TB, 0, stream>>>(dst, dis, E);
  deg_rsqrt_kernel<<<ceil_div(N, TB), TB, 0, stream>>>(dis, N);

  auto conv = [&](const float* hin, const float* W, const float* bb,
                  float* XW, float* accb, int K, int Cout) {
    dim3 gg(ceil_div(ceil_div(N, 16), 4), Cout / 16);
    gemm_wmma_kernel<<<gg, 128, 0, stream>>>(hin, W, XW, N, K, Cout);
    selfloop_init_kernel<<<ceil_div(N * Cout, TB), TB, 0, stream>>>(XW, dis, accb, N, Cout);
    scatter_kernel<<<ceil_div(E * 32, TB), TB, 0, stream>>>(XW, src, dst, dis, accb, E, Cout);
    bias_relu_kernel<<<ceil_div(N * Cout, TB), TB, 0, stream>>>(accb, bb, N, Cout);
  };

  conv(x,    W1, b1, bufA, bufB, F_IN, HID);   // h1 in bufB (N x 128)
  conv(bufB, W2, b2, bufA, bufB, HID,  HID2);  // h2 in bufB (N x 64)
  conv(bufB, W3, b3, bufA, bufB, HID2, OUTC);  // h3 in bufB (N x 32)

  // global mean pool
  pool_count_kernel<<<ceil_div(N, TB), TB, 0, stream>>>(batch, cnt, N);
  pool_sum_kernel<<<ceil_div(N * 32, TB), TB, 0, stream>>>(bufB, batch, out, N);
  pool_div_kernel<<<ceil_div(G * OUTC, TB), TB, 0, stream>>>(out, cnt, G);
}